// TAOBAOGConvGRU_35132832481406
// MI455X (gfx1250) — compile-verified
//
#include <hip/hip_runtime.h>
#include <hip/hip_bf16.h>

typedef __attribute__((ext_vector_type(2))) float v2f;
typedef __attribute__((ext_vector_type(8))) float v8f;

#define NN   50000
#define NE   800000
#define NEL  200000
#define C    128

// workspace layout (float offsets); total ~= 129 MB
#define WS_DEG 0
#define WS_PX  50048
#define WS_PH  (WS_PX + NN * C)
#define WS_Z   (WS_PH + NN * C)
#define WS_HR  (WS_Z + NN * C)
#define WS_P2  (WS_HR + NN * C)
#define WS_BPK (WS_P2 + NN * C)      // 12 * 128 * 128 floats, pair-interleaved weights

#define LROW 132                      // LDS row stride in floats (128 + 4 pad; 132%64==4)

#define WMMA_F32(a, b, c) \
    __builtin_amdgcn_wmma_f32_16x16x4_f32(false, (a), false, (b), (short)0, (c), false, false)

__device__ __forceinline__ void async_load_b128(unsigned lds_off, unsigned long long gaddr) {
    asm volatile("global_load_async_to_lds_b128 %0, %1, off"
                 :: "v"(lds_off), "v"(gaddr) : "memory");
}
__device__ __forceinline__ void wait_async0() {
    asm volatile("s_wait_asynccnt 0x0" ::: "memory");
}

// stage a 32-row x 32-col fp32 slab into LDS buffer (rows clamped to NN-1)
__device__ __forceinline__ void stage_part(const float* __restrict__ src, int m0,
                                           float* buf, int tid) {
    unsigned base = (unsigned)(size_t)(void*)buf;
#pragma unroll
    for (int i = 0; i < 2; ++i) {
        int chunk = i * 128 + tid;            // 256 chunks of 16B
        int row = chunk >> 3;
        int cj  = chunk & 7;
        int grow = m0 + row;
        if (grow > NN - 1) grow = NN - 1;
        unsigned lds = base + (unsigned)(row * LROW + cj * 4) * 4u;
        async_load_b128(lds, (unsigned long long)(size_t)(src + (size_t)grow * C + cj * 4));
    }
}

__global__ __launch_bounds__(256) void zero_f32(float* __restrict__ p, int n) {
    int i = blockIdx.x * 256 + threadIdx.x;
    if (i < n) p[i] = 0.0f;
}

// pack W[k][c] -> Bpk[blk][k/2][c][2] so a B fragment is one aligned b64 load
__global__ __launch_bounds__(256) void pack_weights(const float* __restrict__ Wx0,
                                                    const float* __restrict__ Wx1,
                                                    const float* __restrict__ Wh0,
                                                    const float* __restrict__ Wh1,
                                                    float* __restrict__ Bpk) {
    int tid = blockIdx.x * 256 + threadIdx.x;
    if (tid >= 12 * 16384) return;
    int blk = tid >> 14;                  // part*3 + gate
    int r   = tid & 16383;
    int kk2 = r >> 8;
    int cc  = (r >> 1) & 127;
    int j   = r & 1;
    const float* Ws[4] = {Wx0, Wx1, Wh0, Wh1};
    const float* W = Ws[blk / 3] + (blk % 3) * 16384;
    Bpk[tid] = W[(2 * kk2 + j) * 128 + cc];
}

__global__ __launch_bounds__(256) void deg_kernel(const long long* __restrict__ ei,
                                                  float* __restrict__ deg) {
    int e = blockIdx.x * 256 + threadIdx.x;
    if (e < NE) unsafeAtomicAdd(&deg[ei[e]], 1.0f);
}

__global__ __launch_bounds__(256) void dinv_kernel(float* __restrict__ deg) {
    int i = blockIdx.x * 256 + threadIdx.x;
    if (i < NN) {
        float d = deg[i];
        deg[i] = (d > 0.0f) ? rsqrtf(fmaxf(d, 1e-12f)) : 0.0f;
    }
}

// fused prop(x), prop(H): one wave per edge, lane handles 4 channels
__global__ __launch_bounds__(256) void scatter2_kernel(const float* __restrict__ x,
                                                       const float* __restrict__ H,
                                                       const long long* __restrict__ ei,
                                                       const float* __restrict__ dinv,
                                                       float* __restrict__ px,
                                                       float* __restrict__ pH) {
    int e = blockIdx.x * 8 + (threadIdx.x >> 5);
    if (e >= NE) return;
    int lane = threadIdx.x & 31;
    long long s = ei[e];
    long long d = ei[NE + e];
    float w = -dinv[s] * dinv[d];
    int c = lane * 4;
    float4 xv = *(const float4*)(x + s * C + c);
    float4 hv = *(const float4*)(H + s * C + c);
    float* pxd = px + d * C + c;
    float* pHd = pH + d * C + c;
    unsafeAtomicAdd(pxd + 0, xv.x * w);
    unsafeAtomicAdd(pxd + 1, xv.y * w);
    unsafeAtomicAdd(pxd + 2, xv.z * w);
    unsafeAtomicAdd(pxd + 3, xv.w * w);
    unsafeAtomicAdd(pHd + 0, hv.x * w);
    unsafeAtomicAdd(pHd + 1, hv.y * w);
    unsafeAtomicAdd(pHd + 2, hv.z * w);
    unsafeAtomicAdd(pHd + 3, hv.w * w);
}

__global__ __launch_bounds__(256) void scatter1_kernel(const float* __restrict__ HR,
                                                       const long long* __restrict__ ei,
                                                       const float* __restrict__ dinv,
                                                       float* __restrict__ pHR) {
    int e = blockIdx.x * 8 + (threadIdx.x >> 5);
    if (e >= NE) return;
    int lane = threadIdx.x & 31;
    long long s = ei[e];
    long long d = ei[NE + e];
    float w = -dinv[s] * dinv[d];
    int c = lane * 4;
    float4 hv = *(const float4*)(HR + s * C + c);
    float* pd = pHR + d * C + c;
    unsafeAtomicAdd(pd + 0, hv.x * w);
    unsafeAtomicAdd(pd + 1, hv.y * w);
    unsafeAtomicAdd(pd + 2, hv.z * w);
    unsafeAtomicAdd(pd + 3, hv.w * w);
}

__device__ __forceinline__ float sigmoidf_(float v) { return 1.0f / (1.0f + expf(-v)); }

// Phase 1: gates z, r and the x and H parts of gate h in one fused WMMA GEMM.
// Block = 4 waves, owns 32 rows; A slab async-staged to LDS, double-buffered per part.
// Wave owns cols [w*32, w*32+32) per gate; 12 v8f accumulators (2 row tiles x 2 col tiles x 3 gates).
__global__ __launch_bounds__(128) void phase1_gemm(
    const float* __restrict__ x, const float* __restrict__ px,
    const float* __restrict__ H, const float* __restrict__ pH,
    const float* __restrict__ Bpk,
    const float* __restrict__ bx, const float* __restrict__ bh,
    float* __restrict__ Z, float* __restrict__ HR, float* __restrict__ P2) {
    __shared__ float As[2][32 * LROW];
    const int m0   = blockIdx.x * 32;
    const int tid  = threadIdx.x;
    const int lane = tid & 31;
    const int w    = tid >> 5;
    const int crow = lane & 15;
    const int kh   = lane >> 4;
    const int c0   = w * 32 + crow;
    const int c1   = c0 + 16;

    const float* Asrc[4] = {x, px, H, pH};
    v8f acc[12] = {};

    stage_part(Asrc[0], m0, As[0], tid);
    wait_async0();
    __syncthreads();

#pragma unroll
    for (int p = 0; p < 4; ++p) {
        if (p + 1 < 4) stage_part(Asrc[p + 1], m0, As[(p + 1) & 1], tid);
        const float* Ab = As[p & 1];
        const float* A0 = Ab + crow * LROW + 2 * kh;
        const float* A1 = Ab + (16 + crow) * LROW + 2 * kh;
        for (int kk = 0; kk < C; kk += 4) {
            v2f a0 = *(const v2f*)(A0 + kk);
            v2f a1 = *(const v2f*)(A1 + kk);
#pragma unroll
            for (int g = 0; g < 3; ++g) {
                const float* Bg = Bpk + ((p * 3 + g) << 14) + ((kk >> 1) + kh) * 256;
                v2f b0 = *(const v2f*)(Bg + c0 * 2);
                v2f b1 = *(const v2f*)(Bg + c1 * 2);
                acc[g * 2 + 0]     = WMMA_F32(a0, b0, acc[g * 2 + 0]);
                acc[g * 2 + 1]     = WMMA_F32(a0, b1, acc[g * 2 + 1]);
                acc[6 + g * 2 + 0] = WMMA_F32(a1, b0, acc[6 + g * 2 + 0]);
                acc[6 + g * 2 + 1] = WMMA_F32(a1, b1, acc[6 + g * 2 + 1]);
            }
        }
        if (p + 1 < 4) {
            wait_async0();
            __syncthreads();
        }
    }

#pragma unroll
    for (int rt = 0; rt < 2; ++rt) {
#pragma unroll
        for (int t = 0; t < 2; ++t) {
            const int ch = (t == 0) ? c0 : c1;
            const float bz = bx[ch] + bh[ch];
            const float br = bx[C + ch] + bh[C + ch];
            const float bp = bx[2 * C + ch] + bh[2 * C + ch];
            v8f az = acc[rt * 6 + 0 + t];
            v8f ar = acc[rt * 6 + 2 + t];
            v8f ap = acc[rt * 6 + 4 + t];
#pragma unroll
            for (int i = 0; i < 8; ++i) {
                int row = m0 + rt * 16 + i + 8 * kh;
                if (row > NN - 1) row = NN - 1;          // clamped dup rows: identical writes
                const size_t idx = (size_t)row * C + ch;
                float zv = sigmoidf_(az[i] + bz);
                float rv = sigmoidf_(ar[i] + br);
                Z[idx]  = zv;
                HR[idx] = H[idx] * rv;
                P2[idx] = ap[i] + bp;
            }
        }
    }
}

// Phase 2: Ht = tanh(P2 + [HR|pHR] @ [Wh0[2];Wh1[2]]); h = Z*H + (1-Z)*Ht -> d_out
__global__ __launch_bounds__(128) void phase2_gemm(
    const float* __restrict__ HR, const float* __restrict__ pHR,
    const float* __restrict__ Bpk,
    const float* __restrict__ Z, const float* __restrict__ H,
    const float* __restrict__ P2v, float* __restrict__ hidden) {
    __shared__ float As[2][32 * LROW];
    const int m0   = blockIdx.x * 32;
    const int tid  = threadIdx.x;
    const int lane = tid & 31;
    const int w    = tid >> 5;
    const int crow = lane & 15;
    const int kh   = lane >> 4;
    const int c0   = w * 32 + crow;
    const int c1   = c0 + 16;

    const float* Asrc[2] = {HR, pHR};
    v8f acc[4] = {};

    stage_part(Asrc[0], m0, As[0], tid);
    wait_async0();
    __syncthreads();

#pragma unroll
    for (int p = 0; p < 2; ++p) {
        if (p + 1 < 2) stage_part(Asrc[1], m0, As[1], tid);
        const float* Ab = As[p];
        const float* A0 = Ab + crow * LROW + 2 * kh;
        const float* A1 = Ab + (16 + crow) * LROW + 2 * kh;
        const int blk = (p == 0) ? 8 : 11;            // Wh0[2], Wh1[2] in packed buffer
        const float* Bbase = Bpk + (blk << 14);
        for (int kk = 0; kk < C; kk += 4) {
            v2f a0 = *(const v2f*)(A0 + kk);
            v2f a1 = *(const v2f*)(A1 + kk);
            const float* Bg = Bbase + ((kk >> 1) + kh) * 256;
            v2f b0 = *(const v2f*)(Bg + c0 * 2);
            v2f b1 = *(const v2f*)(Bg + c1 * 2);
            acc[0] = WMMA_F32(a0, b0, acc[0]);
            acc[1] = WMMA_F32(a0, b1, acc[1]);
            acc[2] = WMMA_F32(a1, b0, acc[2]);
            acc[3] = WMMA_F32(a1, b1, acc[3]);
        }
        if (p + 1 < 2) {
            wait_async0();
            __syncthreads();
        }
    }

#pragma unroll
    for (int rt = 0; rt < 2; ++rt) {
#pragma unroll
        for (int t = 0; t < 2; ++t) {
            const int ch = (t == 0) ? c0 : c1;
            v8f a = acc[rt * 2 + t];
#pragma unroll
            for (int i = 0; i < 8; ++i) {
                int row = m0 + rt * 16 + i + 8 * kh;
                if (row > NN - 1) row = NN - 1;
                const size_t idx = (size_t)row * C + ch;
                float ht = tanhf(P2v[idx] + a[i]);
                float zv = Z[idx];
                hidden[idx] = zv * H[idx] + (1.0f - zv) * ht;
            }
        }
    }
}

// scores[p] = sum_c relu(h[s])*relu(h[d]) * (post_w[c,0]+post_w[c,1]) + post_b[0]+post_b[1]
__global__ __launch_bounds__(256) void link_kernel(const float* __restrict__ hidden,
                                                   const long long* __restrict__ eli,
                                                   const float* __restrict__ post_w,
                                                   const float* __restrict__ post_b,
                                                   float* __restrict__ scores) {
    int p = blockIdx.x * 8 + (threadIdx.x >> 5);
    if (p >= NEL) return;
    int lane = threadIdx.x & 31;
    long long s = eli[p];
    long long d = eli[NEL + p];
    int c = lane * 4;
    float4 hs = *(const float4*)(hidden + s * C + c);
    float4 hd = *(const float4*)(hidden + d * C + c);
    float4 wa = *(const float4*)(post_w + 2 * c);
    float4 wb = *(const float4*)(post_w + 2 * c + 4);
    float w0 = wa.x + wa.y, w1 = wa.z + wa.w;
    float w2 = wb.x + wb.y, w3 = wb.z + wb.w;
    float a = fmaxf(hs.x, 0.f) * fmaxf(hd.x, 0.f) * w0 +
              fmaxf(hs.y, 0.f) * fmaxf(hd.y, 0.f) * w1 +
              fmaxf(hs.z, 0.f) * fmaxf(hd.z, 0.f) * w2 +
              fmaxf(hs.w, 0.f) * fmaxf(hd.w, 0.f) * w3;
#pragma unroll
    for (int off = 16; off > 0; off >>= 1) a += __shfl_xor(a, off, 32);
    if (lane == 0) scores[p] = a + post_b[0] + post_b[1];
}

extern "C" void kernel_launch(void* const* d_in, const int* in_sizes, int n_in,
                              void* d_out, int out_size, void* d_ws, size_t ws_size,
                              hipStream_t stream) {
    const float*     x      = (const float*)d_in[0];
    const long long* ei     = (const long long*)d_in[1];
    const long long* eli    = (const long long*)d_in[2];
    const float*     H      = (const float*)d_in[3];
    const float*     Wx0    = (const float*)d_in[4];
    const float*     Wx1    = (const float*)d_in[5];
    const float*     bx     = (const float*)d_in[6];
    const float*     Wh0    = (const float*)d_in[7];
    const float*     Wh1    = (const float*)d_in[8];
    const float*     bh     = (const float*)d_in[9];
    const float*     post_w = (const float*)d_in[10];
    const float*     post_b = (const float*)d_in[11];

    float* out    = (float*)d_out;
    float* scores = out;          // [EL]
    float* hidden = out + NEL;    // [N*C]

    float* ws  = (float*)d_ws;
    float* deg = ws + WS_DEG;
    float* px  = ws + WS_PX;
    float* pH  = ws + WS_PH;
    float* Zb  = ws + WS_Z;
    float* HR  = ws + WS_HR;
    float* P2  = ws + WS_P2;
    float* Bpk = ws + WS_BPK;
    float* pHR = pH;              // pH dead after phase 1; reuse for prop(HR)

    {   // zero deg + px + pH (contiguous)
        int n = WS_Z;
        zero_f32<<<(n + 255) / 256, 256, 0, stream>>>(ws, n);
    }
    pack_weights<<<(12 * 16384) / 256, 256, 0, stream>>>(Wx0, Wx1, Wh0, Wh1, Bpk);
    deg_kernel<<<NE / 256, 256, 0, stream>>>(ei, deg);
    dinv_kernel<<<(NN + 255) / 256, 256, 0, stream>>>(deg);
    scatter2_kernel<<<NE / 8, 256, 0, stream>>>(x, H, ei, deg, px, pH);
    phase1_gemm<<<(NN + 31) / 32, 128, 0, stream>>>(x, px, H, pH, Bpk, bx, bh, Zb, HR, P2);
    zero_f32<<<(NN * C + 255) / 256, 256, 0, stream>>>(pHR, NN * C);
    scatter1_kernel<<<NE / 8, 256, 0, stream>>>(HR, ei, deg, pHR);
    phase2_gemm<<<(NN + 31) / 32, 128, 0, stream>>>(HR, pHR, Bpk, Zb, H, P2, hidden);
    link_kernel<<<(NEL + 7) / 8, 256, 0, stream>>>(hidden, eli, post_w, post_b, scores);
}